// SyntheticGCN_39513699123484
// MI455X (gfx1250) — compile-verified
//
#include <hip/hip_runtime.h>

typedef float v2f __attribute__((ext_vector_type(2)));
typedef float v8f __attribute__((ext_vector_type(8)));

#define D 128
#define NGRAPH 512

// ---------------------------------------------------------------------------
// Degree: deg[dst] += 1 for every edge (deg buffer pre-zeroed via memset)
// ---------------------------------------------------------------------------
__global__ __launch_bounds__(256) void gcn_deg(const long long* __restrict__ dst,
                                               float* __restrict__ deg, int E) {
  int e = blockIdx.x * 256 + threadIdx.x;
  if (e < E) unsafeAtomicAdd(&deg[dst[e]], 1.0f);
}

// dinv[i] = rsqrt(deg[i] + 1)   (in place over the deg buffer)
__global__ __launch_bounds__(256) void gcn_dinv(float* __restrict__ deg, int n) {
  int i = blockIdx.x * 256 + threadIdx.x;
  if (i < n) deg[i] = rsqrtf(deg[i] + 1.0f);
}

// ---------------------------------------------------------------------------
// H = X @ W  with V_WMMA_F32_16X16X4_F32.  One wave per 16x16 output tile;
// block = 8 waves = one 16-row stripe across all 128 columns.
// X tile staged in LDS (A-operand reads are row-strided); W read from global
// (64 KB, L2-resident, lanes 0-15 hit 16 consecutive floats -> coalesced).
// ---------------------------------------------------------------------------
__global__ __launch_bounds__(256) void gcn_gemm_wmma(
    const float* __restrict__ X, const float* __restrict__ W,
    float* __restrict__ H, int n_nodes) {
  __shared__ float sX[16 * D];
  const int tid = threadIdx.x;
  const int row0 = blockIdx.x * 16;

  // cooperative load of the 16x128 X tile (512 float4s / 256 threads)
  {
    const float4* X4 = (const float4*)X;
    float4* sX4 = (float4*)sX;
    for (int i = tid; i < 16 * D / 4; i += 256) {
      int r = i >> 5;  // 32 float4 per row
      int gr = row0 + r;
      float4 v = make_float4(0.f, 0.f, 0.f, 0.f);
      if (gr < n_nodes) v = X4[(size_t)gr * (D / 4) + (i & 31)];
      sX4[i] = v;
    }
  }
  __syncthreads();

  const int wave = tid >> 5;   // 0..7 -> column tile
  const int lane = tid & 31;
  const int half = lane >> 4;  // 0: K pair {0,1}; 1: K pair {2,3}
  const int lr = lane & 15;
  const int n0 = wave * 16;

  v8f c = {};
#pragma unroll
  for (int kk = 0; kk < D; kk += 4) {
    const int ka = kk + 2 * half;
    v2f a, b;
    // A 16x4: lanes 0-15 M=lr, VGPR0=K(ka), VGPR1=K(ka+1)
    a.x = sX[lr * D + ka];
    a.y = sX[lr * D + ka + 1];
    // B 4x16: VGPR0 row ka striped over lanes, VGPR1 row ka+1
    b.x = W[ka * D + n0 + lr];
    b.y = W[(ka + 1) * D + n0 + lr];
    c = __builtin_amdgcn_wmma_f32_16x16x4_f32(false, a, false, b, (short)0, c,
                                              false, false);
  }

// C/D: VGPR r -> row (r + 8*half), col = lr
#pragma unroll
  for (int r = 0; r < 8; ++r) {
    int row = row0 + r + 8 * half;
    if (row < n_nodes) H[(size_t)row * D + n0 + lr] = c[r];
  }
}

// ---------------------------------------------------------------------------
// Edge scatter: agg[dst] += h[src] * dinv[src]*dinv[dst].
// One wave per edge; each lane: float4 gather + 4 hardware f32 atomics (L2).
// ---------------------------------------------------------------------------
__global__ __launch_bounds__(256) void gcn_scatter(
    const long long* __restrict__ src, const long long* __restrict__ dst,
    const float* __restrict__ H, const float* __restrict__ dinv,
    float* __restrict__ agg, int E) {
  int e = (blockIdx.x * 256 + threadIdx.x) >> 5;
  int lane = threadIdx.x & 31;
  if (e >= E) return;
  long long s = src[e], d = dst[e];
  float coef = dinv[s] * dinv[d];
  float4 v = ((const float4*)(H + (size_t)s * D))[lane];
  float* ad = agg + (size_t)d * D + (size_t)lane * 4;
  unsafeAtomicAdd(ad + 0, v.x * coef);
  unsafeAtomicAdd(ad + 1, v.y * coef);
  unsafeAtomicAdd(ad + 2, v.z * coef);
  unsafeAtomicAdd(ad + 3, v.w * coef);
}

// out = relu(agg + h * dinv^2 + b), written in place over agg
__global__ __launch_bounds__(256) void gcn_finalize(
    const float* __restrict__ H, const float* __restrict__ dinv,
    const float* __restrict__ bias, float* __restrict__ agg, int n) {
  int idx = blockIdx.x * 256 + threadIdx.x;
  if (idx >= n * D) return;
  int node = idx >> 7;
  int f = idx & (D - 1);
  float di = dinv[node];
  float v = agg[idx] + H[idx] * di * di + bias[f];
  agg[idx] = fmaxf(v, 0.0f);
}

// segment-sum pool: pooled[batch[node]] += h[node]
__global__ __launch_bounds__(256) void gcn_pool(
    const float* __restrict__ Hf, const long long* __restrict__ batch,
    float* __restrict__ pooled, int n) {
  int idx = blockIdx.x * 256 + threadIdx.x;
  if (idx >= n * D) return;
  int node = idx >> 7;
  int f = idx & (D - 1);
  unsafeAtomicAdd(&pooled[(size_t)batch[node] * D + f], Hf[idx]);
}

// out[g,j] = pooled[g,:] @ Wlin[:,j] + blin[j]   (512 x 2)
__global__ __launch_bounds__(256) void gcn_linear(
    const float* __restrict__ pooled, const float* __restrict__ Wlin,
    const float* __restrict__ blin, float* __restrict__ out) {
  int idx = blockIdx.x * 256 + threadIdx.x;
  if (idx >= NGRAPH * 2) return;
  int g = idx >> 1;
  int j = idx & 1;
  float acc = blin[j];
  for (int f = 0; f < D; ++f) acc += pooled[g * D + f] * Wlin[f * 2 + j];
  out[idx] = acc;
}

extern "C" void kernel_launch(void* const* d_in, const int* in_sizes, int n_in,
                              void* d_out, int out_size, void* d_ws,
                              size_t ws_size, hipStream_t stream) {
  const float* x = (const float*)d_in[0];
  const long long* edge = (const long long*)d_in[1];
  const long long* batch = (const long long*)d_in[2];
  const float* Ws[3] = {(const float*)d_in[3], (const float*)d_in[5],
                        (const float*)d_in[7]};
  const float* bs[3] = {(const float*)d_in[4], (const float*)d_in[6],
                        (const float*)d_in[8]};
  const float* Wlin = (const float*)d_in[9];
  const float* blin = (const float*)d_in[10];

  const int n = in_sizes[0] / D;       // 100000 nodes
  const int E = in_sizes[1] / 2;       // 1600000 edges
  const long long* src = edge;
  const long long* dstp = edge + E;

  const size_t nodeFeat = (size_t)n * D;  // elements
  float* bufA = (float*)d_ws;             // h = x @ W
  float* bufB = bufA + nodeFeat;          // agg / layer output (ping-pong)
  float* dinv = bufB + nodeFeat;          // deg -> dinv in place
  float* pooled = dinv + n;               // 512 x 128

  // degree + dinv (edge weights all 1, so this is layer-invariant)
  hipMemsetAsync(dinv, 0, (size_t)n * sizeof(float), stream);
  gcn_deg<<<(E + 255) / 256, 256, 0, stream>>>(dstp, dinv, E);
  gcn_dinv<<<(n + 255) / 256, 256, 0, stream>>>(dinv, n);

  const float* in = x;
  for (int layer = 0; layer < 3; ++layer) {
    gcn_gemm_wmma<<<(n + 15) / 16, 256, 0, stream>>>(in, Ws[layer], bufA, n);
    hipMemsetAsync(bufB, 0, nodeFeat * sizeof(float), stream);
    gcn_scatter<<<(E + 7) / 8, 256, 0, stream>>>(src, dstp, bufA, dinv, bufB, E);
    gcn_finalize<<<((size_t)n * D + 255) / 256, 256, 0, stream>>>(bufA, dinv,
                                                                  bs[layer],
                                                                  bufB, n);
    in = bufB;
  }

  hipMemsetAsync(pooled, 0, (size_t)NGRAPH * D * sizeof(float), stream);
  gcn_pool<<<((size_t)n * D + 255) / 256, 256, 0, stream>>>(bufB, batch, pooled,
                                                            n);
  gcn_linear<<<(NGRAPH * 2 + 255) / 256, 256, 0, stream>>>(pooled, Wlin, blin,
                                                           (float*)d_out);
}